// GATNodeScorer_43800076485317
// MI455X (gfx1250) — compile-verified
//
#include <hip/hip_runtime.h>

#define NNODES   20000
#define IN_DIM   896
#define CODE_DIM 768
#define NUM_DIM  (IN_DIM - CODE_DIM)   // 128
#define HIDDEN   256
#define HEADS    4
#define CH       64
#define NEG_SLOPE 0.2f
#define CODE_WEIGHT 3.0f

typedef __attribute__((ext_vector_type(2))) float v2f;
typedef __attribute__((ext_vector_type(8))) float v8f;

// ---------------------------------------------------------------------------
// WMMA f32 GEMM, compile-time dims: D[M,256] = op(A)[M,K] @ B[K,256]
// One wave computes a 16x64 output tile (4 WMMA accumulators sharing one
// A fragment per K-step). For the layer-0 input permutation the K loop is
// statically split at k=128 so no per-iteration address/scale selects remain.
// ---------------------------------------------------------------------------
template<int K, int LDA, bool PERMUTE_IN, bool RELU_BIAS>
__global__ __launch_bounds__(256)
void wmma_gemm(const float* __restrict__ A, const float* __restrict__ B,
               const float* __restrict__ bias, float* __restrict__ D, int M)
{
    const int lane = threadIdx.x & 31;
    const int wave = threadIdx.x >> 5;
    const int tid  = blockIdx.x * 8 + wave;
    const int tm   = tid >> 2;          // M tile (16 rows)
    const int tn   = tid & 3;           // N slab (64 cols)
    const int m0   = tm << 4;
    const int n0   = tn << 6;

    const int mrow = lane & 15;         // row of A fragment / col of B fragment
    const int kh   = (lane >> 4) << 1;  // 0 for lanes 0-15, 2 for lanes 16-31

    const float* Arow  = A + (size_t)(m0 + mrow) * LDA;
    const float* Bbase = B + n0 + mrow;

    v8f acc0 = {}, acc1 = {}, acc2 = {}, acc3 = {};

    // One K-step: logical column ka (feeds B), physical A column ka+delta,
    // constant scale (folds away when 1.0f since the lambda is inlined).
    auto kstep = [&](int ka, int delta, float scale) {
        __builtin_prefetch(Arow + ka + delta + 64, 0, 3);
        v2f a;
        a.x = scale * Arow[ka + delta];
        a.y = scale * Arow[ka + delta + 1];

        const float* Bp = Bbase + (size_t)ka * HIDDEN;
        v2f b0, b1, b2, b3;
        b0.x = Bp[0];  b0.y = Bp[HIDDEN +  0];
        b1.x = Bp[16]; b1.y = Bp[HIDDEN + 16];
        b2.x = Bp[32]; b2.y = Bp[HIDDEN + 32];
        b3.x = Bp[48]; b3.y = Bp[HIDDEN + 48];

        acc0 = __builtin_amdgcn_wmma_f32_16x16x4_f32(false, a, false, b0, (short)0, acc0, false, false);
        acc1 = __builtin_amdgcn_wmma_f32_16x16x4_f32(false, a, false, b1, (short)0, acc1, false, false);
        acc2 = __builtin_amdgcn_wmma_f32_16x16x4_f32(false, a, false, b2, (short)0, acc2, false, false);
        acc3 = __builtin_amdgcn_wmma_f32_16x16x4_f32(false, a, false, b3, (short)0, acc3, false, false);
    };

    if (PERMUTE_IN) {
        // k in [0,128): xc[k] = x[768+k]
#pragma unroll 2
        for (int k0 = 0; k0 < NUM_DIM; k0 += 4) kstep(k0 + kh, CODE_DIM, 1.0f);
        // k in [128,K): xc[k] = 3 * x[k-128]
#pragma unroll 2
        for (int k0 = NUM_DIM; k0 < K; k0 += 4) kstep(k0 + kh, -NUM_DIM, CODE_WEIGHT);
    } else {
#pragma unroll 2
        for (int k0 = 0; k0 < K; k0 += 4) kstep(k0 + kh, 0, 1.0f);
    }

    const int rbase = m0 + ((lane >> 4) << 3);   // +8 rows for upper half-wave
    v8f accs[4] = {acc0, acc1, acc2, acc3};
#pragma unroll
    for (int j = 0; j < 4; ++j) {
        const int col = n0 + (j << 4) + mrow;
        float badd = RELU_BIAS ? bias[col] : 0.0f;
#pragma unroll
        for (int v = 0; v < 8; ++v) {
            float d = accs[j][v];
            if (RELU_BIAS) { d += badd; d = fmaxf(d, 0.0f); }
            D[(size_t)(rbase + v) * HIDDEN + col] = d;
        }
    }
}

// ---------------------------------------------------------------------------
__global__ __launch_bounds__(256)
void fill_kernel(float* __restrict__ p, float v, int n)
{
    int i = blockIdx.x * blockDim.x + threadIdx.x;
    if (i < n) p[i] = v;
}

__global__ __launch_bounds__(256)
void copy_kernel(float* __restrict__ dst, const float* __restrict__ src, int n)
{
    int i = blockIdx.x * blockDim.x + threadIdx.x;
    if (i < n) dst[i] = src[i];
}

// hout[dst] += h[src] + rel_emb[etype]*ew   (one thread per edge-channel)
__global__ __launch_bounds__(256)
void rel_agg_kernel(const float* __restrict__ h, const int* __restrict__ ei,
                    const int* __restrict__ etype, const float* __restrict__ ew,
                    const float* __restrict__ rel_emb, float* __restrict__ hout, int E)
{
    long long t = (long long)blockIdx.x * blockDim.x + threadIdx.x;
    if (t >= (long long)E * HIDDEN) return;
    const int e = (int)(t >> 8);
    const int c = (int)(t & (HIDDEN - 1));
    const int s = ei[e], d = ei[E + e];
    const float val = h[(size_t)s * HIDDEN + c] + rel_emb[etype[e] * HIDDEN + c] * ew[e];
    atomicAdd(&hout[(size_t)d * HIDDEN + c], val);
}

// a_src/a_dst[i,head] = sum_c xw[i,head,c] * att_{src,dst}[head,c]
__global__ __launch_bounds__(256)
void att_coef_kernel(const float* __restrict__ xw, const float* __restrict__ att_src,
                     const float* __restrict__ att_dst, float* __restrict__ a_src,
                     float* __restrict__ a_dst, int n)
{
    int t = blockIdx.x * blockDim.x + threadIdx.x;
    if (t >= n * HEADS) return;
    const int i = t >> 2, hd = t & 3;
    const float* xr = xw + (size_t)i * HIDDEN + hd * CH;
    const float* as = att_src + hd * CH;
    const float* ad = att_dst + hd * CH;
    float s = 0.f, d = 0.f;
#pragma unroll 8
    for (int c = 0; c < CH; ++c) { const float v = xr[c]; s += v * as[c]; d += v * ad[c]; }
    a_src[t] = s; a_dst[t] = d;
}

__device__ __forceinline__ void edge_sd(int e, int E, const int* __restrict__ ei,
                                        int& s, int& d)
{
    if (e < E) { s = ei[e]; d = ei[E + e]; }
    else       { s = e - E; d = e - E; }      // self loop
}

__device__ __forceinline__ float lrelu(float x) { return x > 0.f ? x : NEG_SLOPE * x; }

// float atomic max via sign-split (amax initialized to -inf)
__device__ __forceinline__ void atomicMaxF(float* addr, float val)
{
    if (val >= 0.f) atomicMax((int*)addr, __float_as_int(val));
    else            atomicMin((unsigned int*)addr, __float_as_uint(val));
}

__global__ __launch_bounds__(256)
void edge_amax_kernel(const int* __restrict__ ei, const float* __restrict__ a_src,
                      const float* __restrict__ a_dst, float* __restrict__ amax,
                      int E, int n)
{
    int t = blockIdx.x * blockDim.x + threadIdx.x;
    if (t >= (E + n) * HEADS) return;
    const int e = t >> 2, hd = t & 3;
    int s, d; edge_sd(e, E, ei, s, d);
    const float a = lrelu(a_src[s * HEADS + hd] + a_dst[d * HEADS + hd]);
    atomicMaxF(&amax[d * HEADS + hd], a);
}

__global__ __launch_bounds__(256)
void edge_denom_kernel(const int* __restrict__ ei, const float* __restrict__ a_src,
                       const float* __restrict__ a_dst, const float* __restrict__ amax,
                       float* __restrict__ denom, int E, int n)
{
    int t = blockIdx.x * blockDim.x + threadIdx.x;
    if (t >= (E + n) * HEADS) return;
    const int e = t >> 2, hd = t & 3;
    int s, d; edge_sd(e, E, ei, s, d);
    const float a = lrelu(a_src[s * HEADS + hd] + a_dst[d * HEADS + hd]);
    atomicAdd(&denom[d * HEADS + hd], expf(a - amax[d * HEADS + hd]));
}

// out[dst,c] += xw[src,c] * softmax_coef   (one thread per edge-channel)
__global__ __launch_bounds__(256)
void edge_scatter_kernel(const int* __restrict__ ei, const float* __restrict__ a_src,
                         const float* __restrict__ a_dst, const float* __restrict__ amax,
                         const float* __restrict__ denom, const float* __restrict__ xw,
                         float* __restrict__ out, int E, int n)
{
    long long t = (long long)blockIdx.x * blockDim.x + threadIdx.x;
    if (t >= (long long)(E + n) * HIDDEN) return;
    const int e  = (int)(t >> 8);
    const int c  = (int)(t & (HIDDEN - 1));
    const int hd = c >> 6;
    int s, d; edge_sd(e, E, ei, s, d);
    const float a  = lrelu(a_src[s * HEADS + hd] + a_dst[d * HEADS + hd]);
    const float ex = expf(a - amax[d * HEADS + hd]);
    const float coef = ex / denom[d * HEADS + hd];
    atomicAdd(&out[(size_t)d * HIDDEN + c], xw[(size_t)s * HIDDEN + c] * coef);
}

__global__ __launch_bounds__(256)
void bias_add_kernel(float* __restrict__ h, const float* __restrict__ b, int n)
{
    int t = blockIdx.x * blockDim.x + threadIdx.x;
    if (t < n * HIDDEN) h[t] += b[t & (HIDDEN - 1)];
}

// score[i] = dot(h[i,:], Wo) + bo   (one wave per node)
__global__ __launch_bounds__(256)
void score_kernel(const float* __restrict__ h, const float* __restrict__ Wo,
                  const float* __restrict__ bo, float* __restrict__ out, int n)
{
    const int lane = threadIdx.x & 31;
    const int wave = threadIdx.x >> 5;
    const int i = blockIdx.x * 8 + wave;
    if (i >= n) return;
    const float* hr = h + (size_t)i * HIDDEN;
    float s = 0.f;
#pragma unroll
    for (int c = lane; c < HIDDEN; c += 32) s += hr[c] * Wo[c];
#pragma unroll
    for (int off = 16; off; off >>= 1) s += __shfl_down(s, off, 32);
    if (lane == 0) out[i] = s + bo[0];
}

// ---------------------------------------------------------------------------
extern "C" void kernel_launch(void* const* d_in, const int* in_sizes, int n_in,
                              void* d_out, int out_size, void* d_ws, size_t ws_size,
                              hipStream_t stream)
{
    const float* x       = (const float*)d_in[0];
    const int*   ei      = (const int*)  d_in[1];
    const int*   etype   = (const int*)  d_in[2];
    const float* ew      = (const float*)d_in[3];
    const float* rel_emb = (const float*)d_in[4];
    const float* Wp      = (const float*)d_in[5];
    const float* bp      = (const float*)d_in[6];
    const float* W1      = (const float*)d_in[7];
    const float* as1     = (const float*)d_in[8];
    const float* ad1     = (const float*)d_in[9];
    const float* b1      = (const float*)d_in[10];
    const float* W2      = (const float*)d_in[11];
    const float* as2     = (const float*)d_in[12];
    const float* ad2     = (const float*)d_in[13];
    const float* b2      = (const float*)d_in[14];
    const float* Wo      = (const float*)d_in[15];
    const float* bo      = (const float*)d_in[16];

    const int n = in_sizes[0] / IN_DIM;   // 20000
    const int E = in_sizes[2];            // 320000
    const int nelem = n * HIDDEN;

    float* B0    = (float*)d_ws;                 // N x 256
    float* B1    = B0 + (size_t)nelem;           // N x 256
    float* B2    = B1 + (size_t)nelem;           // N x 256
    float* a_src = B2 + (size_t)nelem;           // N x 4
    float* a_dst = a_src + n * HEADS;
    float* amax  = a_dst + n * HEADS;
    float* denom = amax  + n * HEADS;

    const dim3 blk(256);
    const int gemmBlocks = (n / 16) * (HIDDEN / 64) / 8;        // 625
    const int eb  = ((n * HEADS) + 255) / 256;                  // node-head grids
    const int cb  = (nelem + 255) / 256;                        // node-channel grids
    const int ehb = (((E + n) * HEADS) + 255) / 256;            // edge-head grids
    const long long relT = (long long)E * HIDDEN;
    const long long scT  = (long long)(E + n) * HIDDEN;
    const int relB = (int)((relT + 255) / 256);
    const int scB  = (int)((scT + 255) / 256);
    const float NEG_INF = -__builtin_huge_valf();

    // 1) h = relu(xc @ Wp + bp)            -> B0
    wmma_gemm<IN_DIM, IN_DIM, true, true><<<gemmBlocks, blk, 0, stream>>>(x, Wp, bp, B0, n);

    // 2) h = h + segment_sum(h[src] + rel) -> B1
    copy_kernel<<<cb, blk, 0, stream>>>(B1, B0, nelem);
    rel_agg_kernel<<<relB, blk, 0, stream>>>(B0, ei, etype, ew, rel_emb, B1, E);

    // 3) GAT layer 1: in B1, xw B2, out B0
    wmma_gemm<HIDDEN, HIDDEN, false, false><<<gemmBlocks, blk, 0, stream>>>(B1, W1, nullptr, B2, n);
    att_coef_kernel<<<eb, blk, 0, stream>>>(B2, as1, ad1, a_src, a_dst, n);
    fill_kernel<<<eb, blk, 0, stream>>>(amax, NEG_INF, n * HEADS);
    fill_kernel<<<eb, blk, 0, stream>>>(denom, 0.f, n * HEADS);
    fill_kernel<<<cb, blk, 0, stream>>>(B0, 0.f, nelem);
    edge_amax_kernel<<<ehb, blk, 0, stream>>>(ei, a_src, a_dst, amax, E, n);
    edge_denom_kernel<<<ehb, blk, 0, stream>>>(ei, a_src, a_dst, amax, denom, E, n);
    edge_scatter_kernel<<<scB, blk, 0, stream>>>(ei, a_src, a_dst, amax, denom, B2, B0, E, n);
    bias_add_kernel<<<cb, blk, 0, stream>>>(B0, b1, n);

    // 4) GAT layer 2: in B0, xw B2, out B1
    wmma_gemm<HIDDEN, HIDDEN, false, false><<<gemmBlocks, blk, 0, stream>>>(B0, W2, nullptr, B2, n);
    att_coef_kernel<<<eb, blk, 0, stream>>>(B2, as2, ad2, a_src, a_dst, n);
    fill_kernel<<<eb, blk, 0, stream>>>(amax, NEG_INF, n * HEADS);
    fill_kernel<<<eb, blk, 0, stream>>>(denom, 0.f, n * HEADS);
    fill_kernel<<<cb, blk, 0, stream>>>(B1, 0.f, nelem);
    edge_amax_kernel<<<ehb, blk, 0, stream>>>(ei, a_src, a_dst, amax, E, n);
    edge_denom_kernel<<<ehb, blk, 0, stream>>>(ei, a_src, a_dst, amax, denom, E, n);
    edge_scatter_kernel<<<scB, blk, 0, stream>>>(ei, a_src, a_dst, amax, denom, B2, B1, E, n);
    bias_add_kernel<<<cb, blk, 0, stream>>>(B1, b2, n);

    // 5) score = h @ Wo + bo
    score_kernel<<<(n + 7) / 8, blk, 0, stream>>>(B1, Wo, bo, (float*)d_out, n);
}